// SelfSupervisedLoss_25555055412051
// MI455X (gfx1250) — compile-verified
//
#include <hip/hip_runtime.h>
#include <hip/hip_bf16.h>

typedef __attribute__((ext_vector_type(2))) float v2f;
typedef __attribute__((ext_vector_type(8))) float v8f;

namespace ssl {
constexpr int B = 8, S = 512, T = 4, D = 128;
constexpr int M = B * S * T;        // 16384 keys
constexpr int N = B * S * (T - 1);  // 12288 anchors
constexpr int TOT = B * S * T * D;  // 2097152 elements (spikes / mem)
constexpr float INV_TEMP = 10.0f;   // 1 / 0.1
constexpr int REG_BLOCKS = 256;
constexpr int NT = M / 16;          // 1024 column tiles
constexpr int LDS_STRIDE = 132;     // 128 + 4 pad dwords -> conflict-free ds_load_b64
}

// ---------------------------------------------------------------------------
// Kernel 1: per-key inverse norm + diagonal logits sim[i,i]
// ---------------------------------------------------------------------------
__global__ __launch_bounds__(256) void ssl_norm_diag_kernel(
    const float* __restrict__ H, float* __restrict__ invnorm,
    float* __restrict__ diag) {
  using namespace ssl;
  const int m = (blockIdx.x * blockDim.x + threadIdx.x) >> 5;
  const int lane = threadIdx.x & 31;
  if (m >= M) return;

  const float4 k = *((const float4*)(H + (size_t)m * D) + lane);
  float ss = k.x * k.x + k.y * k.y + k.z * k.z + k.w * k.w;
  float dp = 0.0f;
  if (m < N) {
    const int aidx = (m / 3) * 4 + (m % 3);  // anchor flat-hidden index
    const float4 a = *((const float4*)(H + (size_t)aidx * D) + lane);
    dp = a.x * k.x + a.y * k.y + a.z * k.z + a.w * k.w;
  }
#pragma unroll
  for (int off = 16; off > 0; off >>= 1) {
    ss += __shfl_xor(ss, off, 32);
    dp += __shfl_xor(dp, off, 32);
  }
  if (lane == 0) {
    const float inv = 1.0f / fmaxf(sqrtf(ss), 1e-12f);
    invnorm[m] = inv;
    if (m < N) diag[m] = dp * inv * INV_TEMP;
  }
}

// ---------------------------------------------------------------------------
// Kernel 2: fp32-WMMA GEMM + streamed logsumexp.
// Block = 4 waves, each wave owns 16 anchor rows. The 16x128 key tile is
// shared by the whole block: double-buffered in LDS, filled with CDNA5
// async copies (global_load_async_to_lds_b128, ASYNCcnt) so the next tile's
// fill overlaps the current tile's 32 chained v_wmma_f32_16x16x4_f32.
// ---------------------------------------------------------------------------
__global__ __launch_bounds__(128) void ssl_gemm_lse_kernel(
    const float* __restrict__ H, const float* __restrict__ invnorm,
    float* __restrict__ lse) {
  using namespace ssl;
  __shared__ float ktile[2][16 * LDS_STRIDE];  // 2 x 8.25 KB

  const int lane = threadIdx.x & 31;
  const int wave = threadIdx.x >> 5;
  const int row0 = (blockIdx.x * 4 + wave) * 16;
  const int l16 = lane & 15;
  const int hi = lane >> 4;  // 0: K pair {0,1}; 1: K pair {2,3} of 16x4 frag

  // --- preload A fragments (anchor rows are UNnormalized, per reference) ---
  v2f afrag[32];
  {
    const int r = row0 + l16;
    const int aidx = (r / 3) * 4 + (r % 3);
    const float* arow = H + (size_t)aidx * D + 2 * hi;
#pragma unroll
    for (int kb = 0; kb < 32; ++kb)
      afrag[kb] = *(const v2f*)(arow + 4 * kb);
  }

  // issue async copy of key tile starting at row j0 into LDS buffer `buf`
  auto issue_tile = [&](int j0, int buf) {
#pragma unroll
    for (int i = 0; i < 4; ++i) {
      const int e = threadIdx.x + 128 * i;  // float4 index within 16x128 tile
      const int row = e >> 5;               // 32 float4 per key row
      const int c4 = e & 31;
      const unsigned long long gaddr =
          (unsigned long long)(uintptr_t)(H + (size_t)j0 * D + (size_t)e * 4);
      const unsigned int laddr =
          (unsigned int)(uintptr_t)&ktile[buf][row * LDS_STRIDE + c4 * 4];
      asm volatile("global_load_async_to_lds_b128 %0, %1, off"
                   :: "v"(laddr), "v"(gaddr) : "memory");
    }
  };

  float rm[8], rs[8];
#pragma unroll
  for (int e = 0; e < 8; ++e) { rm[e] = -3.0e38f; rs[e] = 0.0f; }

  issue_tile(0, 0);  // prologue: tile 0 -> buffer 0 (4 async ops in flight)

  for (int jt = 0; jt < NT; ++jt) {
    const int cur = jt & 1;
    if (jt + 1 < NT) {
      // back buffer was last read at jt-1; the end-of-iteration barrier of
      // jt-1 guarantees every wave is done with it -> safe to refill now.
      issue_tile((jt + 1) * 16, cur ^ 1);
      asm volatile("s_wait_asynccnt 4" ::: "memory");  // tile `cur` landed
    } else {
      asm volatile("s_wait_asynccnt 0" ::: "memory");
    }
    __syncthreads();  // tile `cur` visible to all 4 waves

    const float* kbase = &ktile[cur][l16 * LDS_STRIDE + 2 * hi];
    v8f c0 = {}, c1 = {};  // two independent WMMA chains (even/odd K blocks)
#pragma unroll
    for (int kb = 0; kb < 32; kb += 2) {
      const v2f b0 = *(const v2f*)(kbase + 4 * kb);
      const v2f b1 = *(const v2f*)(kbase + 4 * (kb + 1));
      c0 = __builtin_amdgcn_wmma_f32_16x16x4_f32(false, afrag[kb], false, b0,
                                                 (short)0, c0, false, false);
      c1 = __builtin_amdgcn_wmma_f32_16x16x4_f32(false, afrag[kb + 1], false,
                                                 b1, (short)0, c1, false,
                                                 false);
    }

    // each lane's 8 accumulators share one key column -> lane-scalar scale
    const float scale = invnorm[jt * 16 + l16] * INV_TEMP;
#pragma unroll
    for (int e = 0; e < 8; ++e) {
      const float v = (c0[e] + c1[e]) * scale;
      const float nm = fmaxf(rm[e], v);
      rs[e] = rs[e] * __expf(rm[e] - nm) + __expf(v - nm);
      rm[e] = nm;
    }
    __syncthreads();  // all waves done reading tile `cur` before its refill
  }

  // merge (max,sum) across the 16 lanes of each half (rows 0-7 / 8-15)
#pragma unroll
  for (int off = 1; off < 16; off <<= 1) {
#pragma unroll
    for (int e = 0; e < 8; ++e) {
      const float om = __shfl_xor(rm[e], off, 32);
      const float os = __shfl_xor(rs[e], off, 32);
      const float nm = fmaxf(rm[e], om);
      rs[e] = rs[e] * __expf(rm[e] - nm) + os * __expf(om - nm);
      rm[e] = nm;
    }
  }

  if (l16 == 0) {  // lane 0 -> rows 0..7, lane 16 -> rows 8..15
#pragma unroll
    for (int e = 0; e < 8; ++e)
      lse[row0 + hi * 8 + e] = rm[e] + __logf(rs[e]);
  }
}

// ---------------------------------------------------------------------------
// Kernel 3: deterministic per-block partial sums for spikes and mem^2
// ---------------------------------------------------------------------------
__global__ __launch_bounds__(256) void ssl_reg_partial_kernel(
    const float* __restrict__ spikes, const float* __restrict__ mem,
    float* __restrict__ pspike, float* __restrict__ pmem) {
  using namespace ssl;
  const float4* s4 = (const float4*)spikes;
  const float4* m4 = (const float4*)mem;
  const int n4 = TOT / 4;
  float s = 0.0f, q = 0.0f;
  for (int i = blockIdx.x * blockDim.x + threadIdx.x; i < n4;
       i += gridDim.x * blockDim.x) {
    const float4 a = s4[i];
    s += (a.x + a.y) + (a.z + a.w);
    const float4 b = m4[i];
    q += (b.x * b.x + b.y * b.y) + (b.z * b.z + b.w * b.w);
  }
  __shared__ float sh[2][8];
  const int lane = threadIdx.x & 31, wv = threadIdx.x >> 5;
#pragma unroll
  for (int off = 16; off > 0; off >>= 1) {
    s += __shfl_xor(s, off, 32);
    q += __shfl_xor(q, off, 32);
  }
  if (lane == 0) { sh[0][wv] = s; sh[1][wv] = q; }
  __syncthreads();
  if (threadIdx.x == 0) {
    float ts = 0.0f, tq = 0.0f;
    for (int w = 0; w < 8; ++w) { ts += sh[0][w]; tq += sh[1][w]; }
    pspike[blockIdx.x] = ts;
    pmem[blockIdx.x] = tq;
  }
}

// ---------------------------------------------------------------------------
// Kernel 4: single-block finalize -> 5 outputs
// ---------------------------------------------------------------------------
__global__ __launch_bounds__(256) void ssl_finalize_kernel(
    const float* __restrict__ lse, const float* __restrict__ diag,
    const int* __restrict__ targets, const float* __restrict__ pspike,
    const float* __restrict__ pmem, float* __restrict__ out) {
  using namespace ssl;
  float cs = 0.0f, cnt = 0.0f, ps = 0.0f, pq = 0.0f;
  for (int i = threadIdx.x; i < N; i += blockDim.x) {
    const float v = (targets[i / 3] != 0) ? 1.0f : 0.0f;
    cs += (lse[i] - diag[i]) * v;
    cnt += v;
  }
  for (int i = threadIdx.x; i < REG_BLOCKS; i += blockDim.x) {
    ps += pspike[i];
    pq += pmem[i];
  }
  __shared__ float sh[4][8];
  const int lane = threadIdx.x & 31, wv = threadIdx.x >> 5;
#pragma unroll
  for (int off = 16; off > 0; off >>= 1) {
    cs += __shfl_xor(cs, off, 32);
    cnt += __shfl_xor(cnt, off, 32);
    ps += __shfl_xor(ps, off, 32);
    pq += __shfl_xor(pq, off, 32);
  }
  if (lane == 0) { sh[0][wv] = cs; sh[1][wv] = cnt; sh[2][wv] = ps; sh[3][wv] = pq; }
  __syncthreads();
  if (threadIdx.x == 0) {
    float a = 0, b = 0, c = 0, d = 0;
    for (int w = 0; w < 8; ++w) {
      a += sh[0][w]; b += sh[1][w]; c += sh[2][w]; d += sh[3][w];
    }
    const float tcl = a / fmaxf(b, 1.0f);
    const float rate = c / (float)TOT;
    const float sreg = (rate - 0.02f) * (rate - 0.02f);
    const float mreg = d / (float)TOT;
    out[0] = tcl + 0.01f * sreg + 0.001f * mreg;
    out[1] = tcl;
    out[2] = sreg;
    out[3] = mreg;
    out[4] = rate;
  }
}

// ---------------------------------------------------------------------------
extern "C" void kernel_launch(void* const* d_in, const int* in_sizes, int n_in,
                              void* d_out, int out_size, void* d_ws,
                              size_t ws_size, hipStream_t stream) {
  using namespace ssl;
  const float* H       = (const float*)d_in[0];  // full_hiddens [B,S,T,D]
  const int*   targets = (const int*)d_in[1];    // [B,S]
  const float* spikes  = (const float*)d_in[2];  // [B,S,T,D]
  const float* mem     = (const float*)d_in[3];  // [B,S,T,D]
  float* out = (float*)d_out;

  float* ws      = (float*)d_ws;
  float* invnorm = ws;                  // M      = 16384
  float* diag    = invnorm + M;         // N      = 12288
  float* lse     = diag + N;            // N      = 12288
  float* pspike  = lse + N;             // 256
  float* pmem    = pspike + REG_BLOCKS; // 256

  // 1. per-key inverse norms + diagonal logits (1 wave / row, 8 waves / block)
  ssl_norm_diag_kernel<<<M / 8, 256, 0, stream>>>(H, invnorm, diag);
  // 2. fp32-WMMA GEMM with async-LDS double buffering and streamed logsumexp
  ssl_gemm_lse_kernel<<<(N / 16) / 4, 128, 0, stream>>>(H, invnorm, lse);
  // 3. spike / mem partial sums (deterministic tree, no atomics)
  ssl_reg_partial_kernel<<<REG_BLOCKS, 256, 0, stream>>>(spikes, mem, pspike, pmem);
  // 4. finalize the 5 scalar outputs
  ssl_finalize_kernel<<<1, 256, 0, stream>>>(lse, diag, targets, pspike, pmem, out);
}